// CapsuleLayer_81003083203581
// MI455X (gfx1250) — compile-verified
//
#include <hip/hip_runtime.h>
#include <hip/hip_bf16.h>

// ---- problem constants (from reference) ----
#define B_SZ    64      // batch
#define K_CAPS  32      // capsules
#define R_SZ    2048    // routes
#define I_CH    16      // in channels (GEMM K)
#define O_CH    32      // out channels (GEMM N)
#define ITERS   5
#define R_CHUNK 128
#define N_CHUNKS (R_SZ / R_CHUNK)   // 16

typedef __attribute__((ext_vector_type(2))) float v2f;
typedef __attribute__((ext_vector_type(8))) float v8f;
typedef __attribute__((ext_vector_type(4))) int   v4i;

#define AS1 __attribute__((address_space(1)))
#define AS3 __attribute__((address_space(3)))

#ifndef __has_builtin
#define __has_builtin(x) 0
#endif
#if __has_builtin(__builtin_amdgcn_global_load_async_to_lds_b128)
#define ASYNC_LDS 1
#else
#define ASYNC_LDS 0
#endif
#if __has_builtin(__builtin_amdgcn_s_wait_asynccnt)
#define WAIT_ASYNC0() __builtin_amdgcn_s_wait_asynccnt(0)
#else
#define WAIT_ASYNC0() asm volatile("s_wait_asynccnt 0x0" ::: "memory")
#endif

__device__ __forceinline__ float bf2f(unsigned int h16) {
    union { unsigned int u; float f; } c;
    c.u = h16 << 16;
    return c.f;
}

// =====================================================================
// Kernel 1: u_hat[b,k,r,o] = sum_i x[b,r,i] * W[k,r,i,o]
// Per (k,r): [64x16]x[16x32] GEMM with V_WMMA_F32_16X16X4_F32.
// Workgroup = (r, octet of 8 k's); one wave per k; x[:,r,:] staged in LDS.
// Result bounced through a wave-private LDS tile so global stores are
// 16-byte b128 (instead of scattered 2-byte b16).
// =====================================================================
__global__ __launch_bounds__(256)
void uhat_gemm_kernel(const float* __restrict__ x,
                      const float* __restrict__ W,
                      __hip_bfloat16* __restrict__ u_hat) {
    const int octet = blockIdx.x;          // 0..3
    const int r     = blockIdx.y;          // 0..2047
    const int tid   = threadIdx.x;
    const int wave  = tid >> 5;
    const int lane  = tid & 31;
    const int ml    = lane & 15;           // row/col within half-wave
    const int hi    = lane >> 4;           // 0 = lanes 0-15, 1 = lanes 16-31
    const int k     = octet * 8 + wave;    // capsule handled by this wave

    __shared__ float xs[B_SZ * I_CH];                    // 4 KB  [b][i]
    __shared__ __hip_bfloat16 st[8][B_SZ * O_CH];        // 32 KB wave-private tiles

    for (int e = tid; e < B_SZ * I_CH; e += 256) {
        const int bb = e >> 4, i = e & 15;
        xs[e] = x[((size_t)bb * R_SZ + r) * I_CH + i];
    }
    __syncthreads();

    const float* Wkr = W + ((size_t)k * R_SZ + r) * (I_CH * O_CH); // [i][o] 16x32

    v8f acc[4][2];
    #pragma unroll
    for (int mt = 0; mt < 4; ++mt)
        #pragma unroll
        for (int nt = 0; nt < 2; ++nt)
            #pragma unroll
            for (int j = 0; j < 8; ++j) acc[mt][nt][j] = 0.0f;

    #pragma unroll
    for (int kk = 0; kk < 4; ++kk) {       // 4 K-steps of 4 over I_CH=16
        const int i0 = kk * 4;
        // A operand (16x4 f32): lanes<16 -> M=ml, K={i0,i0+1}; lanes>=16 -> K={i0+2,i0+3}
        v2f a[4];
        #pragma unroll
        for (int mt = 0; mt < 4; ++mt) {
            const float* xp = &xs[(mt * 16 + ml) * I_CH + i0 + 2 * hi];
            a[mt].x = xp[0];
            a[mt].y = xp[1];
        }
        // B operand (4x16 f32): VGPR0 = row K{i0 | i0+2}, VGPR1 = row K{i0+1 | i0+3}
        v2f bb[2];
        #pragma unroll
        for (int nt = 0; nt < 2; ++nt) {
            const float* wp = Wkr + (size_t)(i0 + 2 * hi) * O_CH + nt * 16 + ml;
            bb[nt].x = wp[0];
            bb[nt].y = wp[O_CH];
        }
        #pragma unroll
        for (int mt = 0; mt < 4; ++mt)
            #pragma unroll
            for (int nt = 0; nt < 2; ++nt)
                acc[mt][nt] = __builtin_amdgcn_wmma_f32_16x16x4_f32(
                    false, a[mt], false, bb[nt], (short)0, acc[mt][nt],
                    false, false);
    }

    // D layout: VGPR j holds M=j (lanes 0-15) / M=j+8 (lanes 16-31), N=ml.
    // Transpose through wave-private LDS (in-order per wave, no barrier).
    #pragma unroll
    for (int mt = 0; mt < 4; ++mt)
        #pragma unroll
        for (int nt = 0; nt < 2; ++nt)
            #pragma unroll
            for (int j = 0; j < 8; ++j) {
                const int m = mt * 16 + j + 8 * hi;   // batch index
                const int o = nt * 16 + ml;
                st[wave][m * O_CH + o] = __float2bfloat16(acc[mt][nt][j]);
            }

    // coalesced write-back: 8 x b128 stores per lane
    const size_t kbase = ((size_t)k * R_SZ + r) * O_CH;   // [b][k][r][o] minus b term
    const uint4* stv = (const uint4*)st[wave];            // 256 groups of 8 bf16
    for (int g = lane; g < (B_SZ * O_CH) / 8; g += 32) {
        const int b  = g >> 2;         // 4 groups (64 B) per batch row
        const int og = (g & 3) * 8;
        *(uint4*)(u_hat + (size_t)b * (K_CAPS * R_SZ * O_CH) + kbase + og) = stv[g];
    }
}

// =====================================================================
// Kernel 2 (x5): one fused routing iteration over an r-chunk.
// Workgroup = (b, chunk of 128 r's), all 32 capsules resident:
//   - stage u_hat[b, :, chunk, :] (256 KB bf16) in LDS via async-to-LDS
//   - b_t = b_{t-1} + sum_o u*v_{t-1}; c = softmax_k(b_t)
//   - s_partial[b,chunk,k,o] = sum_{r in chunk} c * u   (no atomics)
// =====================================================================
__global__ __launch_bounds__(256)
void routing_pass_kernel(const __hip_bfloat16* __restrict__ u_hat,
                         const float* __restrict__ v_prev,
                         float* __restrict__ b_ij,
                         float* __restrict__ s_partial,
                         int iter) {
    const int chunk = blockIdx.x;   // 0..15
    const int b     = blockIdx.y;   // 0..63
    const int tid   = threadIdx.x;
    const int r0    = chunk * R_CHUNK;

    extern __shared__ __align__(16) char smem[];
    __hip_bfloat16* u_lds = (__hip_bfloat16*)smem;                       // 256 KB [k][rr][o]
    float* c_lds = (float*)(smem + (size_t)K_CAPS * R_CHUNK * O_CH * 2); // 16 KB [rr][k]
    float* v_lds = c_lds + R_CHUNK * K_CAPS;                             // 4 KB [k][o]

    // ---- stage u_hat chunk: 16384 x 16B transfers ----
    const int ngroups = K_CAPS * R_CHUNK * O_CH / 8;  // groups of 8 bf16
    uint4* u_lds_v = (uint4*)u_lds;
    for (int g = tid; g < ngroups; g += 256) {
        const int k      = g >> 9;      // 512 groups per capsule
        const int within = g & 511;
        const uint4* src =
            (const uint4*)(u_hat + (((size_t)b * K_CAPS + k) * R_SZ + r0) * O_CH)
            + within;
#if ASYNC_LDS
        __builtin_amdgcn_global_load_async_to_lds_b128(
            (AS1 v4i*)src, (AS3 v4i*)&u_lds_v[g], 0, 0);
#else
        u_lds_v[g] = *src;
#endif
    }
    if (iter > 0) {
        for (int e = tid; e < K_CAPS * O_CH; e += 256)
            v_lds[e] = v_prev[(size_t)b * K_CAPS * O_CH + e];
    }
#if ASYNC_LDS
    WAIT_ASYNC0();
#endif
    __syncthreads();

    // ---- phase 1: logits + softmax over k, one thread per r ----
    if (tid < R_CHUNK) {
        const int rr = tid;
        float* brow = b_ij + ((size_t)b * R_SZ + (r0 + rr)) * K_CAPS;
        float logit[K_CAPS];
        for (int k = 0; k < K_CAPS; ++k) {
            float acc = 0.0f;
            if (iter > 0) {
                const uint4* up = (const uint4*)&u_lds[((size_t)k * R_CHUNK + rr) * O_CH];
                const float* vp = &v_lds[k * O_CH];
                #pragma unroll
                for (int q = 0; q < 4; ++q) {        // 4 x b128 = 32 bf16
                    const uint4 w = up[q];
                    const unsigned int ws[4] = {w.x, w.y, w.z, w.w};
                    #pragma unroll
                    for (int d = 0; d < 4; ++d) {
                        acc += bf2f(ws[d] & 0xffffu) * vp[q * 8 + d * 2 + 0];
                        acc += bf2f(ws[d] >> 16)     * vp[q * 8 + d * 2 + 1];
                    }
                }
                acc += brow[k];
            }
            logit[k] = acc;
            if (iter < ITERS - 1) brow[k] = acc;   // b_t for next iteration
        }
        float m = logit[0];
        for (int k = 1; k < K_CAPS; ++k) m = fmaxf(m, logit[k]);
        float sum = 0.0f;
        for (int k = 0; k < K_CAPS; ++k) {
            logit[k] = __expf(logit[k] - m);
            sum += logit[k];
        }
        const float inv = 1.0f / sum;
        for (int k = 0; k < K_CAPS; ++k)
            c_lds[rr * K_CAPS + k] = logit[k] * inv;
    }
    __syncthreads();

    // ---- phase 2: partial s; thread owns (k, 4 outs), serial over rr ----
    const int k  = tid >> 3;
    const int og = (tid & 7) * 4;
    float a0 = 0.f, a1 = 0.f, a2 = 0.f, a3 = 0.f;
    for (int rr = 0; rr < R_CHUNK; ++rr) {
        const float c = c_lds[rr * K_CAPS + k];
        const uint2 w = *(const uint2*)&u_lds[((size_t)k * R_CHUNK + rr) * O_CH + og];
        a0 += c * bf2f(w.x & 0xffffu);
        a1 += c * bf2f(w.x >> 16);
        a2 += c * bf2f(w.y & 0xffffu);
        a3 += c * bf2f(w.y >> 16);
    }
    float* sp = s_partial + (((size_t)b * N_CHUNKS + chunk) * K_CAPS + k) * O_CH + og;
    sp[0] = a0; sp[1] = a1; sp[2] = a2; sp[3] = a3;
}

// =====================================================================
// Kernel 3 (x5): deterministic chunk-sum + squash. wave <-> (b,k), lane <-> o.
// =====================================================================
__global__ __launch_bounds__(1024)
void squash_kernel(const float* __restrict__ s_partial,
                   float* __restrict__ v_out) {
    const int b = blockIdx.x;
    const int k = threadIdx.x >> 5;
    const int o = threadIdx.x & 31;
    float s = 0.0f;
    #pragma unroll
    for (int c = 0; c < N_CHUNKS; ++c)
        s += s_partial[(((size_t)b * N_CHUNKS + c) * K_CAPS + k) * O_CH + o];
    float n2 = s * s;
    #pragma unroll
    for (int off = 16; off >= 1; off >>= 1)
        n2 += __shfl_xor(n2, off, 32);
    const float norm  = sqrtf(n2);
    const float scale = n2 / ((1.0f + n2) * (norm + 1e-8f));
    v_out[((size_t)b * K_CAPS + k) * O_CH + o] = scale * s;
}

// =====================================================================
extern "C" void kernel_launch(void* const* d_in, const int* in_sizes, int n_in,
                              void* d_out, int out_size, void* d_ws, size_t ws_size,
                              hipStream_t stream) {
    (void)in_sizes; (void)n_in; (void)out_size; (void)ws_size;
    const float* x = (const float*)d_in[0];                 // [B,R,I]
    const float* W = (const float*)d_in[1];                 // [K,R,I,O]
    float* out = (float*)d_out;                             // [B,K,O]

    // workspace layout (bytes)
    char* ws = (char*)d_ws;
    const size_t UHAT_BYTES = (size_t)B_SZ * K_CAPS * R_SZ * O_CH * 2;      // 256 MiB
    const size_t BIJ_BYTES  = (size_t)B_SZ * R_SZ * K_CAPS * 4;             // 16 MiB
    const size_t SP_BYTES   = (size_t)B_SZ * N_CHUNKS * K_CAPS * O_CH * 4;  // 4 MiB
    __hip_bfloat16* u_hat   = (__hip_bfloat16*)ws;
    float* b_ij      = (float*)(ws + UHAT_BYTES);
    float* s_partial = (float*)(ws + UHAT_BYTES + BIJ_BYTES);
    float* v_buf     = (float*)(ws + UHAT_BYTES + BIJ_BYTES + SP_BYTES);

    const size_t lds_bytes = (size_t)K_CAPS * R_CHUNK * O_CH * 2   // u_hat chunk
                           + (size_t)R_CHUNK * K_CAPS * 4          // c
                           + (size_t)K_CAPS * O_CH * 4;            // v
    (void)hipFuncSetAttribute((const void*)routing_pass_kernel,
                              hipFuncAttributeMaxDynamicSharedMemorySize,
                              (int)lds_bytes);

    uhat_gemm_kernel<<<dim3(4, R_SZ), 256, 0, stream>>>(x, W, u_hat);

    for (int t = 0; t < ITERS; ++t) {
        routing_pass_kernel<<<dim3(N_CHUNKS, B_SZ), 256, lds_bytes, stream>>>(
            u_hat, v_buf, b_ij, s_partial, t);
        squash_kernel<<<B_SZ, 1024, 0, stream>>>(
            s_partial, (t == ITERS - 1) ? out : v_buf);
    }
}